// SparseCoupleDeConvTest_26963804684451
// MI455X (gfx1250) — compile-verified
//
#include <hip/hip_runtime.h>

// ---------------- problem constants (match reference) ----------------
#define DD   41
#define HH   159
#define WW   159
#define MD   20
#define MH   79
#define MW   79
#define CIN  32
#define COUT 64
#define NPTS 150000            // B * N_PER_BATCH
#define NGRP (NPTS / 16)       // 9375, exact
#define HWSZ (HH * WW)         // 25281
#define DHW  (DD * HH * WW)    // 1036521
#define MIDVOX (2 * MD * MH * MW)          // 249640 voxels
#define MIDFLOATS ((long)MIDVOX * COUT)    // 15,976,960 f32
#define OUTFLOATS ((long)2 * CIN * DHW)    // 66,337,344 f32

// workspace layout (in floats)
#define ZROW_OFF  MIDFLOATS                 // 64-float always-zero row
#define W1P_OFF   (MIDFLOATS + 64)          // packed w1: 27*16*64 float2 = 55296 f32
#define W2P_OFF   (W1P_OFF + 55296)         // packed w2: 27*32*32 float2 = 55296 f32
#define NPAIR     27648                     // float2 entries in each packed weight

typedef float v2f __attribute__((ext_vector_type(2)));
typedef float v4f __attribute__((ext_vector_type(4)));
typedef float v8f __attribute__((ext_vector_type(8)));

// ---------------- zero fill (vectorized, grid-stride) ----------------
template <bool NT>
__global__ void zero_f32(float* __restrict__ p, long n4) {
  long i = (long)blockIdx.x * blockDim.x + threadIdx.x;
  long stride = (long)gridDim.x * blockDim.x;
  v4f z = {0.f, 0.f, 0.f, 0.f};
  for (long j = i; j < n4; j += stride) {
    if (NT) __builtin_nontemporal_store(z, (v4f*)p + j);
    else    ((v4f*)p)[j] = z;
  }
}

// ---------------- one-shot weight repack into WMMA B-operand order -------
// w1p entry e=(k*16 + kr/2)*64 + col : {w1[k][kr][col], w1[k][kr+1][col]}
// w2p entry e=(k*32 + co/2)*32 + ci  : {w2[k][ci][co],  w2[k][ci][co+1]}
__global__ void pack_weights(const float* __restrict__ w1,
                             const float* __restrict__ w2,
                             float* __restrict__ w1p,
                             float* __restrict__ w2p) {
  int tid = blockIdx.x * blockDim.x + threadIdx.x;
  if (tid >= NPAIR) return;
  int k = tid >> 10, rem = tid & 1023;
  {
    int pr = rem >> 6, col = rem & 63, kr = pr * 2;
    const float* wk = w1 + k * (CIN * COUT);
    v2f v = {wk[kr * COUT + col], wk[(kr + 1) * COUT + col]};
    ((v2f*)w1p)[tid] = v;
  }
  {
    int pr = rem >> 5, col = rem & 31, kr = pr * 2;
    const float* wk = w2 + k * (CIN * COUT);
    v2f v = {wk[col * COUT + kr], wk[col * COUT + kr + 1]};
    ((v2f*)w2p)[tid] = v;
  }
}

// valid stride-2 tap: d = p - k must be >=0, even, and q = d/2 in range
__device__ __forceinline__ bool tap_ok(int dz, int dy, int dx) {
  return ((dz | dy | dx) >= 0) && (((dz | dy | dx) & 1) == 0) &&
         (dz <= 2 * (MD - 1)) && (dy <= 2 * (MH - 1)) && (dx <= 2 * (MW - 1));
}

// ---------------- stage 1: sparse scatter conv (stride 2) ----------------
// One wave32 per 16-point group.  A = features[16x32], B = w1[k][32x64].
// D tiles [16x16] atomically scattered into mid[voxel][64].
__global__ void scatter_conv(const float* __restrict__ feat,
                             const int*  __restrict__ coors,
                             const float* __restrict__ w1p,
                             float* __restrict__ mid) {
  int wave = (int)((blockIdx.x * blockDim.x + threadIdx.x) >> 5);
  if (wave >= NGRP) return;                      // wave-uniform exit
  int lane = threadIdx.x & 31;
  int r16  = lane & 15;                          // my row within the tile
  int hl   = lane >> 4;                          // half-wave selector
  int row  = wave * 16 + r16;

  int4 c = ((const int4*)coors)[row];            // (b, z, y, x)

  // A regs, 8 K-steps (K=32): lanes0-15 hold k={4s,4s+1}, lanes16-31 k={4s+2,4s+3}
  v2f a[8];
  const float* frow = feat + (long)row * CIN + hl * 2;
#pragma unroll
  for (int s = 0; s < 8; ++s) a[s] = *(const v2f*)(frow + 4 * s);

#pragma unroll 1
  for (int kz = 0; kz < 3; ++kz)
#pragma unroll 1
  for (int ky = 0; ky < 3; ++ky)
#pragma unroll 1
  for (int kx = 0; kx < 3; ++kx) {
    int dz = c.y - kz, dy = c.z - ky, dx = c.w - kx;
    bool ok = tap_ok(dz, dy, dx);
    int vIdx = ok ? (((c.x * MD + (dz >> 1)) * MH + (dy >> 1)) * MW + (dx >> 1)) : -1;
    if (__ballot(ok) == 0ull) continue;          // wave-uniform tap skip

    int kofs = (kz * 3 + ky) * 3 + kx;
    // packed B base for this tap + half-wave: float2 entry (kofs*16 + 2s+hl)*64 + col
    const float* wb = w1p + 2l * ((kofs * 16 + hl) * 64);
#pragma unroll 1
    for (int t = 0; t < 4; ++t) {                // 4 N-tiles of COUT=64
      int col = t * 16 + r16;
      v8f acc = {0.f, 0.f, 0.f, 0.f, 0.f, 0.f, 0.f, 0.f};
#pragma unroll
      for (int s = 0; s < 8; ++s) {
        v2f b = *(const v2f*)(wb + 2 * (2 * s * 64 + col));   // one b64, coalesced
        acc = __builtin_amdgcn_wmma_f32_16x16x4_f32(
            false, a[s], false, b, (short)0, acc, false, false);
      }
      // D vgpr j: lanes0-15 = row j, lanes16-31 = row j+8 ; channel = col
#pragma unroll
      for (int j = 0; j < 8; ++j) {
        int m  = j + hl * 8;
        int vi = __shfl(vIdx, m, 32);            // coherent per half-wave
        if (vi >= 0) atomicAdd(&mid[(long)vi * COUT + col], acc[j]);
      }
    }
  }
}

// ---------------- stage 2: sparse gather transposed conv ----------------
// A = gathered mid rows [16x64], B = w2[k]^T [64x32]; accumulate over taps,
// then store 32 channels per point into (B, CIN, D, H, W) output.
__global__ void gather_deconv(const int*  __restrict__ coors,
                              const float* __restrict__ w2p,
                              const float* __restrict__ mid,
                              float* __restrict__ out) {
  int wave = (int)((blockIdx.x * blockDim.x + threadIdx.x) >> 5);
  if (wave >= NGRP) return;
  int lane = threadIdx.x & 31;
  int r16  = lane & 15;
  int hl   = lane >> 4;
  int row  = wave * 16 + r16;

  int4 c = ((const int4*)coors)[row];            // (b, z, y, x)

  v8f acc0 = {0.f,0.f,0.f,0.f,0.f,0.f,0.f,0.f};  // ci 0..15
  v8f acc1 = {0.f,0.f,0.f,0.f,0.f,0.f,0.f,0.f};  // ci 16..31

#pragma unroll 1
  for (int kz = 0; kz < 3; ++kz)
#pragma unroll 1
  for (int ky = 0; ky < 3; ++ky)
#pragma unroll 1
  for (int kx = 0; kx < 3; ++kx) {
    int dz = c.y - kz, dy = c.z - ky, dx = c.w - kx;
    bool ok = tap_ok(dz, dy, dx);
    if (__ballot(ok) == 0ull) continue;
    int vIdx = ok ? (((c.x * MD + (dz >> 1)) * MH + (dy >> 1)) * MW + (dx >> 1)) : -1;

    // A: my row of mid (64 floats); invalid rows read the persistent zero row
    long rbase = ok ? (long)vIdx * COUT : (long)ZROW_OFF;
    const float* mrow = mid + rbase + hl * 2;
    v2f a[16];
#pragma unroll
    for (int s = 0; s < 16; ++s) a[s] = *(const v2f*)(mrow + 4 * s);

    int kofs = (kz * 3 + ky) * 3 + kx;
    const float* wb = w2p + 2l * ((kofs * 32 + hl) * 32);
#pragma unroll 1
    for (int t = 0; t < 2; ++t) {                // 2 N-tiles of CIN=32
      int col = t * 16 + r16;                    // ci
      v8f acc = (t == 0) ? acc0 : acc1;
#pragma unroll
      for (int s = 0; s < 16; ++s) {
        v2f b = *(const v2f*)(wb + 2 * (2 * s * 32 + col));   // one b64, coalesced
        acc = __builtin_amdgcn_wmma_f32_16x16x4_f32(
            false, a[s], false, b, (short)0, acc, false, false);
      }
      if (t == 0) acc0 = acc; else acc1 = acc;
    }
  }

  // store: out[((b*CIN+ci)*DD+z)*HH*WW + y*WW + x]   (streaming, non-temporal)
  int base = c.x * CIN * DHW + c.y * HWSZ + c.z * WW + c.w;
#pragma unroll
  for (int j = 0; j < 8; ++j) {
    int m  = j + hl * 8;                         // row held in D vgpr j
    int bo = __shfl(base, m, 32);
    __builtin_nontemporal_store(acc0[j], out + (long)bo + (long)r16 * DHW);
    __builtin_nontemporal_store(acc1[j], out + (long)bo + (long)(r16 + 16) * DHW);
  }
}

// ---------------- launcher ----------------
extern "C" void kernel_launch(void* const* d_in, const int* in_sizes, int n_in,
                              void* d_out, int out_size, void* d_ws, size_t ws_size,
                              hipStream_t stream) {
  const float* feat  = (const float*)d_in[0];
  const int*   coors = (const int*)d_in[1];
  // d_in[2] = batch_size (compile-time constant 2 here)
  const float* w1    = (const float*)d_in[3];
  const float* w2    = (const float*)d_in[4];
  float* out = (float*)d_out;
  float* ws  = (float*)d_ws;
  float* mid = ws;                               // [0, MIDFLOATS) + zero row
  float* w1p = ws + W1P_OFF;
  float* w2p = ws + W2P_OFF;

  // zero mid + zero-row (L2-resident, regular stores) and out (streaming, NT)
  zero_f32<false><<<2048, 256, 0, stream>>>(mid, (MIDFLOATS + 64) / 4);
  zero_f32<true ><<<4096, 256, 0, stream>>>(out, OUTFLOATS / 4);
  pack_weights<<<(NPAIR + 255) / 256, 256, 0, stream>>>(w1, w2, w1p, w2p);

  const int WAVES_PER_BLOCK = 8;                 // 256 threads = 8 wave32
  int blocks = (NGRP + WAVES_PER_BLOCK - 1) / WAVES_PER_BLOCK;
  scatter_conv<<<blocks, 256, 0, stream>>>(feat, coors, w1p, mid);
  gather_deconv<<<blocks, 256, 0, stream>>>(coors, w2p, mid, out);
}